// ParameterizedExperts_9672266350753
// MI455X (gfx1250) — compile-verified
//
#include <hip/hip_runtime.h>
#include <cstddef>

// Problem constants (match reference)
constexpr int E     = 8;
constexpr int T     = 8192;
constexpr int D_IN  = 2048;
constexpr int D_OUT = 4096;

// Tiling
constexpr int BM  = 128;         // token rows per block
constexpr int BN  = 128;         // out-features per block
constexpr int BK  = 32;          // K slice per LDS stage
constexpr int PAD = 4;           // row padding (floats): stride 36 floats = 144B (16B aligned)
constexpr int LD  = BK + PAD;    // 36
constexpr int ABUF = BM * LD;    // floats per A buffer
constexpr int BBUF = BN * LD;    // floats per B buffer

typedef __attribute__((ext_vector_type(2))) float v2f;
typedef __attribute__((ext_vector_type(8))) float v8f;
typedef __attribute__((ext_vector_type(4))) int   v4i;

typedef __attribute__((address_space(1))) v4i gv4i;  // global 16B chunk
typedef __attribute__((address_space(3))) v4i lv4i;  // LDS 16B chunk

// ---- CDNA5 async global->LDS copy of 16 bytes (ASYNCcnt-tracked), with fallback ----
__device__ __forceinline__ void async_copy16(const float* gsrc, float* ldst) {
#if __has_builtin(__builtin_amdgcn_global_load_async_to_lds_b128)
    __builtin_amdgcn_global_load_async_to_lds_b128(
        (gv4i*)gsrc, (lv4i*)ldst, 0, 0);
#else
    *(float4*)ldst = *(const float4*)gsrc;
#endif
}

__device__ __forceinline__ void wait_async0() {
#if __has_builtin(__builtin_amdgcn_s_wait_asynccnt)
    __builtin_amdgcn_s_wait_asynccnt(0);
#else
    asm volatile("s_wait_asynccnt 0x0" ::: "memory");
#endif
}

__global__ __launch_bounds__(256)
void moe_grouped_gemm_f32_wmma(const float* __restrict__ x,
                               const float* __restrict__ w,
                               const int*   __restrict__ expert_freq,
                               float*       __restrict__ out) {
    // ---- map blockIdx.x -> (expert, row tile) via per-expert tile prefix sums ----
    int tile = blockIdx.x;
    int found_e = -1, row_tile = 0, row_off = 0, cnt = 0;
    {
        int acc = 0, off = 0;
        #pragma unroll
        for (int i = 0; i < E; ++i) {
            int c  = expert_freq[i];
            int nt = (c + BM - 1) / BM;
            if (found_e < 0 && tile < acc + nt) {
                found_e  = i;
                row_tile = tile - acc;
                row_off  = off;
                cnt      = c;
            }
            acc += nt;
            off += c;
        }
    }
    if (found_e < 0) return;   // block-uniform exit (EXEC all-1s below)

    const int m_base = row_off + row_tile * BM;
    int rows_valid   = cnt - row_tile * BM;
    if (rows_valid > BM) rows_valid = BM;
    const int n_base = blockIdx.y * BN;

    const size_t w_expert = (size_t)found_e * D_OUT * D_IN;

    // double-buffered LDS: 2 * (18KB + 18KB) = 72KB of the 320KB WGP pool
    __shared__ float sA[2 * ABUF];
    __shared__ float sB[2 * BBUF];

    const int tid  = threadIdx.x;
    const int lane = tid & 31;
    const int wave = tid >> 5;       // 0..7
    const int half = lane >> 4;      // lane half: selects K pair / M+8
    const int l16  = lane & 15;

    const int wm = (wave >> 1) * 32; // wave's M region: 0,32,64,96 (two 16-row subtiles)
    const int wn = (wave & 1) * 64;  // wave's N half: 0 or 64 (four 16-col subtiles)

    // 2x4 accumulator tiles of 16x16 f32 (64 VGPRs)
    v8f c00 = {}, c01 = {}, c02 = {}, c03 = {};
    v8f c10 = {}, c11 = {}, c12 = {}, c13 = {};

    // ---- stage one BK-slice into LDS buffer `buf` (async) ----
    auto stage = [&](int kt, int buf) {
        const int aoff = buf * ABUF;
        const int boff = buf * BBUF;
        // A: BM x BK = 1024 16B chunks, 4 per thread
        #pragma unroll
        for (int i = 0; i < 4; ++i) {
            int lin = tid + i * 256;
            int r   = lin >> 3;          // 0..127
            int c4  = lin & 7;           // 0..7
            int gr  = m_base + r;
            if (gr >= T) gr = T - 1;     // clamp: ragged rows masked at store
            async_copy16(x + (size_t)gr * D_IN + kt + c4 * 4,
                         &sA[aoff + r * LD + c4 * 4]);
        }
        // B: BN x BK = 1024 16B chunks, 4 per thread
        #pragma unroll
        for (int i = 0; i < 4; ++i) {
            int lin = tid + i * 256;
            int r   = lin >> 3;
            int c4  = lin & 7;
            async_copy16(w + w_expert + (size_t)(n_base + r) * D_IN + kt + c4 * 4,
                         &sB[boff + r * LD + c4 * 4]);
        }
    };

    // ---- pipeline prologue ----
    int p = 0;
    stage(0, 0);
    wait_async0();
    __syncthreads();

    for (int kt = 0; kt < D_IN; kt += BK) {
        if (kt + BK < D_IN) stage(kt + BK, p ^ 1);   // prefetch next slice async

        const int aoff = p * ABUF;
        const int boff = p * BBUF;
        // ---- compute: 8 k-steps, 8 WMMAs each -> 64 WMMAs per K-iter ----
        #pragma unroll
        for (int k0 = 0; k0 < BK; k0 += 4) {
            const int ka = k0 + 2 * half;
            v2f a0 = *(const v2f*)&sA[aoff + (wm +  0 + l16) * LD + ka];
            v2f a1 = *(const v2f*)&sA[aoff + (wm + 16 + l16) * LD + ka];
            v2f b0 = *(const v2f*)&sB[boff + (wn +  0 + l16) * LD + ka];
            v2f b1 = *(const v2f*)&sB[boff + (wn + 16 + l16) * LD + ka];
            v2f b2 = *(const v2f*)&sB[boff + (wn + 32 + l16) * LD + ka];
            v2f b3 = *(const v2f*)&sB[boff + (wn + 48 + l16) * LD + ka];
            c00 = __builtin_amdgcn_wmma_f32_16x16x4_f32(false, a0, false, b0, (short)0, c00, false, false);
            c01 = __builtin_amdgcn_wmma_f32_16x16x4_f32(false, a0, false, b1, (short)0, c01, false, false);
            c02 = __builtin_amdgcn_wmma_f32_16x16x4_f32(false, a0, false, b2, (short)0, c02, false, false);
            c03 = __builtin_amdgcn_wmma_f32_16x16x4_f32(false, a0, false, b3, (short)0, c03, false, false);
            c10 = __builtin_amdgcn_wmma_f32_16x16x4_f32(false, a1, false, b0, (short)0, c10, false, false);
            c11 = __builtin_amdgcn_wmma_f32_16x16x4_f32(false, a1, false, b1, (short)0, c11, false, false);
            c12 = __builtin_amdgcn_wmma_f32_16x16x4_f32(false, a1, false, b2, (short)0, c12, false, false);
            c13 = __builtin_amdgcn_wmma_f32_16x16x4_f32(false, a1, false, b3, (short)0, c13, false, false);
        }

        wait_async0();      // next slice fully in LDS
        __syncthreads();    // all waves done reading buf p / writing buf p^1
        p ^= 1;
    }

    // ---- store C: VGPR i of a tile holds rows (base+i | base+i+8) split by lane half ----
    const int col = n_base + wn + l16;
    #pragma unroll
    for (int i = 0; i < 8; ++i) {
        int rl0 = wm + i + 8 * half;          // tiles c0x
        if (rl0 < rows_valid) {
            float* orow = out + (size_t)(m_base + rl0) * D_OUT;
            orow[col +  0] = c00[i];
            orow[col + 16] = c01[i];
            orow[col + 32] = c02[i];
            orow[col + 48] = c03[i];
        }
        int rl1 = wm + 16 + i + 8 * half;     // tiles c1x
        if (rl1 < rows_valid) {
            float* orow = out + (size_t)(m_base + rl1) * D_OUT;
            orow[col +  0] = c10[i];
            orow[col + 16] = c11[i];
            orow[col + 32] = c12[i];
            orow[col + 48] = c13[i];
        }
    }
}

extern "C" void kernel_launch(void* const* d_in, const int* in_sizes, int n_in,
                              void* d_out, int out_size, void* d_ws, size_t ws_size,
                              hipStream_t stream) {
    const float* x    = (const float*)d_in[0];
    const float* w    = (const float*)d_in[1];
    const int*   freq = (const int*)d_in[2];
    float*       out  = (float*)d_out;

    dim3 grid(T / BM + E, D_OUT / BN);   // static upper bound on row tiles (graph-capture safe)
    dim3 block(256);
    moe_grouped_gemm_f32_wmma<<<grid, block, 0, stream>>>(x, w, freq, out);
}